// CRF_24103356465443
// MI455X (gfx1250) — compile-verified
//
#include <hip/hip_runtime.h>
#include <hip/hip_bf16.h>
#include <stddef.h>

// Problem dims (fixed by the reference)
#define B_  64
#define L_  1024
#define D_  1024
#define C_  128            // NUM_CLASS + 2
#define M_  (B_ * L_)      // 65536 rows of the emission GEMM
#define STOP_ (C_ - 1)

typedef __attribute__((ext_vector_type(2))) float v2f;
typedef __attribute__((ext_vector_type(8))) float v8f;

// ---------------------------------------------------------------------------
// Kernel 1: emis[m, c] = x[m, :] @ W[:, c] + bias[c]   (f32, WMMA 16x16x4)
// One wave per 16x16 output tile; 8 waves per block cover all N=128 columns
// for one 16-row M tile, so a block re-reads its 16 x-rows from cache 8x.
// A-frag layout (16x4 f32): lanes 0-15 hold K=0,1 (VGPR0,1); lanes 16-31 K=2,3.
// B-frag layout (4x16 f32): VGPR0 lanes0-15=K0,row; lanes16-31=K2; VGPR1 K1/K3.
// ---------------------------------------------------------------------------
__global__ __launch_bounds__(256) void crf_emis_gemm(
    const float* __restrict__ x,      // [M_, D_]
    const float* __restrict__ W,      // [D_, C_]
    const float* __restrict__ bias,   // [C_]
    float* __restrict__ emis)         // [M_, C_]
{
  const int wave = threadIdx.x >> 5;        // 0..7 -> N tile
  const int lane = threadIdx.x & 31;
  const int half = lane >> 4;               // 0: K 0/1, 1: K 2/3
  const int lrow = lane & 15;
  const int m0 = blockIdx.x * 16;
  const int n0 = wave * 16;
  const int col = n0 + lrow;

  // A: x[m0+lrow][k + 2*half + {0,1}]  (8B-aligned pair -> b64 load)
  const float* arow = x + (size_t)(m0 + lrow) * D_ + 2 * half;
  // B: W[k + 2*half + {0,1}][col]
  const float* bcol = W + (size_t)(2 * half) * C_ + col;

  v8f acc = {};
#pragma unroll 4
  for (int k = 0; k < D_; k += 4) {
    v2f a = *(const v2f*)(arow + k);
    v2f bb;
    bb.x = bcol[(size_t)k * C_];
    bb.y = bcol[(size_t)(k + 1) * C_];
    acc = __builtin_amdgcn_wmma_f32_16x16x4_f32(
        /*neg_a=*/false, a, /*neg_b=*/false, bb,
        /*c_mod=*/(short)0, acc, /*reuse_a=*/false, /*reuse_b=*/false);
  }

  // D layout: VGPR r -> M = r (lanes 0-15) / r+8 (lanes 16-31), N = lrow
  const float bv = bias[col];
  float* orow = emis + (size_t)(m0 + 8 * half) * C_ + col;
#pragma unroll
  for (int r = 0; r < 8; ++r)
    orow[(size_t)r * C_] = acc[r] + bv;
}

// ---------------------------------------------------------------------------
// Kernel 2: Viterbi forward scan + backtrack. One block per batch element,
// 128 threads (thread c owns tag c). Transposed transitions staged in LDS
// with a 130-float row stride (bank index = (2c + p) % 64 -> conflict-free
// across a wave). Scores live in LDS (uniform-address broadcast reads).
// ---------------------------------------------------------------------------
__global__ __launch_bounds__(128) void crf_viterbi(
    const float* __restrict__ emis,          // [B_, L_, C_]
    const unsigned char* __restrict__ mask,  // [B_, L_] bool
    const float* __restrict__ T,             // [C_, C_]
    unsigned char* __restrict__ bp,          // [L_, B_, C_]
    float* __restrict__ out)                 // [B_, L_] (tags as f32)
{
  __shared__ float Tt[C_ * 130];   // Tt[c*130 + p] = T[p][c]
  __shared__ float sc[C_];
  __shared__ float red_v[C_];
  __shared__ int   red_i[C_];

  const int b = blockIdx.x;
  const int c = threadIdx.x;

  // Stage transposed transitions (coalesced: fixed p, consecutive c per iter).
  for (int p = 0; p < C_; ++p)
    Tt[c * 130 + p] = T[p * C_ + c];
  sc[c] = 0.0f;

  // seq_len = sum(mask[b, :])
  int cnt = 0;
  for (int i = c; i < L_; i += C_)
    cnt += (mask[(size_t)b * L_ + i] != 0) ? 1 : 0;
  red_i[c] = cnt;
  __syncthreads();
  for (int s = 64; s > 0; s >>= 1) {
    if (c < s) red_i[c] += red_i[c + s];
    __syncthreads();
  }
  const int seq_len = red_i[0];
  __syncthreads();

  // ---- forward scan over time ----
  for (int l = 0; l < L_; ++l) {
    float best = -3.402823e38f;
    int bi = 0;
#pragma unroll 4
    for (int p = 0; p < C_; ++p) {
      float v = sc[p] + Tt[c * 130 + p];
      if (v > best) { best = v; bi = p; }   // strict > keeps first argmax
    }
    bp[(size_t)l * (B_ * C_) + (size_t)b * C_ + c] = (unsigned char)bi;

    const float e  = emis[((size_t)b * L_ + l) * C_ + c];
    const bool  ml = mask[(size_t)b * L_ + l] != 0;
    const float ns = best + e;
    __syncthreads();                 // all reads of sc[] for this step done
    if (ml) sc[c] = ns;              // masked update (prefix mask)
    __syncthreads();
  }

  // ---- final scores + block argmax ----
  red_v[c] = sc[c] + T[c * C_ + STOP_];
  red_i[c] = c;
  __syncthreads();
  for (int s = 64; s > 0; s >>= 1) {
    if (c < s) {
      if (red_v[c + s] > red_v[c]) {
        red_v[c] = red_v[c + s];
        red_i[c] = red_i[c + s];
      }
    }
    __syncthreads();
  }

  // ---- backtracking (serial pointer chase, one thread per batch) ----
  if (c == 0) {
    int cur = red_i[0];
    for (int l = L_ - 1; l >= 0; --l) {
      out[(size_t)b * L_ + l] = (l <= seq_len - 1) ? (float)cur : 0.0f;
      if (l >= 1 && l <= seq_len - 1)
        cur = bp[(size_t)l * (B_ * C_) + (size_t)b * C_ + cur];
    }
  }
}

// ---------------------------------------------------------------------------
extern "C" void kernel_launch(void* const* d_in, const int* in_sizes, int n_in,
                              void* d_out, int out_size, void* d_ws, size_t ws_size,
                              hipStream_t stream) {
  (void)in_sizes; (void)n_in; (void)out_size; (void)ws_size;
  const float*         x    = (const float*)d_in[0];
  const unsigned char* mask = (const unsigned char*)d_in[1];  // bool [B,L]
  /* d_in[2] = y : unused by the reference output */
  const float*         W    = (const float*)d_in[3];
  const float*         bias = (const float*)d_in[4];
  const float*         Tr   = (const float*)d_in[5];
  float*               out  = (float*)d_out;

  float*         emis = (float*)d_ws;                               // 32 MB
  unsigned char* bp   = (unsigned char*)d_ws + (size_t)M_ * C_ * 4; // +8 MB

  crf_emis_gemm<<<M_ / 16, 256, 0, stream>>>(x, W, bias, emis);
  crf_viterbi<<<B_, C_, 0, stream>>>(emis, mask, Tr, bp, out);
}